// PINNLayer_27977416966567
// MI455X (gfx1250) — compile-verified
//
#include <hip/hip_runtime.h>

#define N_NODES 100000
#define N_EDGES 3200000
#define HUMAN_EXHALATION_FLOW 0.0052f
#define EDGE_ITER 4   // 3,200,000 edges = 3125 blocks * 256 threads * 4 iters

typedef __attribute__((ext_vector_type(2))) float v2f;
typedef __attribute__((ext_vector_type(8))) float v8f;

// -------- zero the node accumulator region of d_out (replay-safe) --------
__global__ __launch_bounds__(256) void zero_acc_kernel(float* __restrict__ out) {
    int i = blockIdx.x * blockDim.x + threadIdx.x;
    if (i < N_NODES) out[i] = 0.0f;
}

// -------- per-edge: conv via V_WMMA_F32_16X16X4_F32 + scatter + out2 --------
// Wave handles 32 edges: tile0 = e0..e0+15, tile1 = e0+16..e0+31.
// A = weights broadcast over M (lane half h holds K = 2h, 2h+1).
// B = flow values (lane column n = lane&15, rows K = 2h, 2h+1).
// D has identical rows -> lane L reads edge (e0+L)'s value from acc[0].
__global__ __launch_bounds__(256) void edge_kernel(
    const float* __restrict__ origin,      // N_NODES x 16 x 3
    const float* __restrict__ flow,        // (N_EDGES+2) x 3 x 4
    const float* __restrict__ conv_w,      // 1 x 4 x 3 x 3
    const float* __restrict__ conv_b,      // 1
    const int*   __restrict__ edge_index,  // 2 x N_EDGES
    float* out)                            // [0,N_NODES) acc | out2 rows
{
    const int tid  = blockIdx.x * blockDim.x + threadIdx.x;
    const int lane = threadIdx.x & 31;
    const int h    = lane >> 4;   // lane half -> K offset 2h
    const int n    = lane & 15;   // B column (edge within tile)
    const int gw   = tid >> 5;    // global wave id

    // A-matrix weights: flat weight index t = kh*12 + kw*4 + c maps to
    // conv_w flat index c*9 + kh*3 + kw. Lane needs t = 4j + 2h + v.
    float aw0[9], aw1[9];
#pragma unroll
    for (int j = 0; j < 9; ++j) {
#pragma unroll
        for (int v = 0; v < 2; ++v) {
            int t   = 4 * j + 2 * h + v;
            int kh  = t / 12;
            int rem = t - kh * 12;
            int kw  = rem >> 2;
            int c   = rem & 3;
            float w = conv_w[c * 9 + kh * 3 + kw];
            if (v == 0) aw0[j] = w; else aw1[j] = w;
        }
    }
    const float bias = conv_b[0];

    float* acc_nodes = out;             // atomic accumulation region
    float* out2      = out + N_NODES;   // [src, dst, val] rows

#pragma unroll 1
    for (int it = 0; it < EDGE_ITER; ++it) {
        const long long e0 = ((long long)gw * EDGE_ITER + it) * 32;

        const float* f0 = flow + (e0 + n)      * 12 + 2 * h;
        const float* f1 = flow + (e0 + 16 + n) * 12 + 2 * h;
        __builtin_prefetch(f0 + 32 * 12, 0, 1);   // next iteration's span
        __builtin_prefetch(f1 + 32 * 12, 0, 1);

        v2f b0[9], b1[9];
#pragma unroll
        for (int j = 0; j < 9; ++j) {
            b0[j] = *(const v2f*)(f0 + 4 * j);
            b1[j] = *(const v2f*)(f1 + 4 * j);
        }

        v8f acc0 = {bias, bias, bias, bias, bias, bias, bias, bias};
        v8f acc1 = acc0;
#pragma unroll
        for (int j = 0; j < 9; ++j) {
            v2f a = {aw0[j], aw1[j]};
            acc0 = __builtin_amdgcn_wmma_f32_16x16x4_f32(
                false, a, false, b0[j], (short)0, acc0, false, false);
            acc1 = __builtin_amdgcn_wmma_f32_16x16x4_f32(
                false, a, false, b1[j], (short)0, acc1, false, false);
        }

        const long long e   = e0 + lane;          // this lane's edge
        const float     val = (h == 0) ? acc0[0] : acc1[0];

        const int src = edge_index[e];
        const int dst = edge_index[N_EDGES + e];

        out2[e * 3 + 0] = (float)src;
        out2[e * 3 + 1] = (float)dst;
        out2[e * 3 + 2] = val;

        if (src != dst) {
            const float conc_s = origin[(long long)src * 48 + 45];
            const float size_s = origin[(long long)src * 48 + 47];
            const float size_d = origin[(long long)dst * 48 + 47];
            const float msg    = val * conc_s;
            atomicAdd(&acc_nodes[src], -msg / size_s);
            atomicAdd(&acc_nodes[dst],  msg / size_d);
        }
    }
}

// -------- per-node finalize (in place over the accumulator) --------
__global__ __launch_bounds__(256) void node_kernel(
    const float* __restrict__ origin, float* __restrict__ out)
{
    int i = blockIdx.x * blockDim.x + threadIdx.x;
    if (i >= N_NODES) return;
    float conc   = origin[(long long)i * 48 + 45];
    float people = origin[(long long)i * 48 + 46];
    float size   = origin[(long long)i * 48 + 47];
    float nn     = out[i];
    float exh    = HUMAN_EXHALATION_FLOW * people / size;
    float mask   = (i == N_NODES - 1) ? 0.0f : 1.0f;
    out[i] = conc + (nn + exh) * mask;
}

extern "C" void kernel_launch(void* const* d_in, const int* in_sizes, int n_in,
                              void* d_out, int out_size, void* d_ws, size_t ws_size,
                              hipStream_t stream) {
    const float* origin     = (const float*)d_in[0];
    const float* flow       = (const float*)d_in[1];
    const float* conv_w     = (const float*)d_in[2];
    const float* conv_b     = (const float*)d_in[3];
    const int*   edge_index = (const int*)d_in[4];
    float* out = (float*)d_out;

    (void)d_ws; (void)ws_size; (void)in_sizes; (void)n_in; (void)out_size;

    zero_acc_kernel<<<(N_NODES + 255) / 256, 256, 0, stream>>>(out);
    edge_kernel<<<N_EDGES / (256 * EDGE_ITER), 256, 0, stream>>>(
        origin, flow, conv_w, conv_b, edge_index, out);
    node_kernel<<<(N_NODES + 255) / 256, 256, 0, stream>>>(origin, out);
}